// MT2VEncoder_49469433315657
// MI455X (gfx1250) — compile-verified
//
#include <hip/hip_runtime.h>
#include <math.h>

#define LSER 1024
#define NB 4
#define ND 8
#define NM (NB*ND)            // 32 (b,d) matrices
#define IMG 224
#define OUTN (NB*2*ND*IMG*IMG)
#define EPSN 1e-8f
#define CLIPM 1e-6f

typedef __attribute__((ext_vector_type(2))) float v2f;
typedef __attribute__((ext_vector_type(8))) float v8f;

// ---- monotone float<->uint keys so atomicMin/Max on u32 give float min/max ----
__device__ __forceinline__ unsigned fkey(float f) {
  unsigned u = __float_as_uint(f);
  return (u & 0x80000000u) ? ~u : (u | 0x80000000u);
}
__device__ __forceinline__ float kfloat(unsigned k) {
  unsigned u = (k & 0x80000000u) ? (k & 0x7FFFFFFFu) : ~k;
  return __uint_as_float(u);
}

// ---------------- init reduction slots (must reset every call) ----------------
__global__ void initKernel(unsigned* gminKey, unsigned* gmaxKey, unsigned* xKeys) {
  int tid = threadIdx.x;
  if (tid < NM) { gminKey[tid] = 0xFFFFFFFFu; gmaxKey[tid] = 0u; }
  if (tid == 0) { xKeys[0] = 0xFFFFFFFFu; xKeys[1] = 0u; }
}

// ---------------- global min/max of x (32768 elements) ----------------
__global__ __launch_bounds__(1024) void reduceXKernel(const float* __restrict__ x,
                                                      unsigned* __restrict__ xKeys) {
  __shared__ float mn[1024], mx[1024];
  int tid = threadIdx.x;
  float v = x[blockIdx.x * 1024 + tid];
  mn[tid] = v; mx[tid] = v;
  __syncthreads();
  for (int off = 512; off > 0; off >>= 1) {
    if (tid < off) {
      mn[tid] = fminf(mn[tid], mn[tid + off]);
      mx[tid] = fmaxf(mx[tid], mx[tid + off]);
    }
    __syncthreads();
  }
  if (tid == 0) {
    atomicMin(&xKeys[0], fkey(mn[0]));
    atomicMax(&xKeys[1], fkey(mx[0]));
  }
}

// ---- prep: c = cos(theta) = clipped xnorm, s = sin(theta) = sqrt(1-c^2),
//      st = raw series, all transposed to [b,d,l] ----
__global__ void prepKernel(const float* __restrict__ x, const unsigned* __restrict__ xKeys,
                           float* __restrict__ c, float* __restrict__ s,
                           float* __restrict__ st) {
  int t = blockIdx.x * 256 + threadIdx.x;          // 0..32767
  float lo = kfloat(xKeys[0]);
  float hi = kfloat(xKeys[1]);
  float rng = hi - lo;
  int b = t >> 13;           // / (8*1024)
  int d = (t >> 10) & 7;
  int l = t & (LSER - 1);
  float val = x[((b << 10) + l) * ND + d];
  float n = (rng < EPSN) ? 0.0f : (val - lo) / (rng + EPSN);
  float u = n * 2.0f - 1.0f;
  u = fminf(fmaxf(u, -1.0f + CLIPM), 1.0f - CLIPM);
  c[t]  = u;
  s[t]  = sqrtf(fmaxf(1.0f - u * u, 0.0f));
  st[t] = val;
}

// ---- per-(b,d) GAF min/max via V_WMMA_F32_16X16X4_F32 rank-2 GEMM tiles ----
// A (16x4): lanes 0-15 hold M=lane with (K0,K1)=(c_i,s_i); lanes 16-31 (K2,K3)=0
// B (4x16): lanes 0-15 hold N=lane with (K0,K1)=(c_j,-s_j); lanes 16-31 = 0
__global__ __launch_bounds__(256) void gafMinMaxKernel(const float* __restrict__ c,
                                                       const float* __restrict__ s,
                                                       unsigned* __restrict__ gminKey,
                                                       unsigned* __restrict__ gmaxKey) {
  int m = blockIdx.y;                          // matrix (b*8+d)
  const float* cm = c + m * LSER;
  const float* sm = s + m * LSER;
  int wave = threadIdx.x >> 5;                 // 0..7
  int lane = threadIdx.x & 31;
  int half = lane >> 4;                        // 0 => lanes 0-15
  int l16  = lane & 15;
  int ti   = blockIdx.x * 8 + wave;            // row tile 0..63

  float cv = cm[ti * 16 + l16];
  float sv = sm[ti * 16 + l16];
  v2f a;
  a.x = half ? 0.0f : cv;
  a.y = half ? 0.0f : sv;

  float vmin = 3.0e38f, vmax = -3.0e38f;
  for (int tj = 0; tj < 64; ++tj) {
    float cj = cm[tj * 16 + l16];
    float sj = sm[tj * 16 + l16];
    v2f bv;
    bv.x = half ? 0.0f : cj;
    bv.y = half ? 0.0f : -sj;
    v8f dacc = {};
#if __has_builtin(__builtin_amdgcn_wmma_f32_16x16x4_f32)
    dacc = __builtin_amdgcn_wmma_f32_16x16x4_f32(false, a, false, bv,
                                                 (short)0, dacc, false, false);
#else
    // fallback (should not be used on gfx1250): scalar rank-2 product of own row
    #pragma unroll
    for (int q = 0; q < 8; ++q) dacc[q] = a.x * bv.x + a.y * bv.y;
#endif
    #pragma unroll
    for (int q = 0; q < 8; ++q) {
      float v = dacc[q];
      vmin = fminf(vmin, v);
      vmax = fmaxf(vmax, v);
    }
  }
  __shared__ float smn[256], smx[256];
  smn[threadIdx.x] = vmin; smx[threadIdx.x] = vmax;
  __syncthreads();
  for (int off = 128; off > 0; off >>= 1) {
    if (threadIdx.x < off) {
      smn[threadIdx.x] = fminf(smn[threadIdx.x], smn[threadIdx.x + off]);
      smx[threadIdx.x] = fmaxf(smx[threadIdx.x], smx[threadIdx.x + off]);
    }
    __syncthreads();
  }
  if (threadIdx.x == 0) {
    atomicMin(&gminKey[m], fkey(smn[0]));
    atomicMax(&gmaxKey[m], fkey(smx[0]));
  }
}

// count of j with |si - sj| <= t against sorted LDS array (two binary searches)
__device__ __forceinline__ int pairCount(const float* ss, float si, float t) {
  float hv = si + t, lv = si - t;
  int lo = 0, hi = LSER;
  while (lo < hi) { int mid = (lo + hi) >> 1; if (ss[mid] <= hv) lo = mid + 1; else hi = mid; }
  int ub = lo;
  lo = 0; hi = LSER;
  while (lo < hi) { int mid = (lo + hi) >> 1; if (ss[mid] < lv) lo = mid + 1; else hi = mid; }
  return ub - lo;
}

// ---- RP quantile threshold per (b,d): bitonic sort + bit-bisection selection ----
__global__ __launch_bounds__(1024) void rpThrKernel(const float* __restrict__ st,
                                                    float* __restrict__ thr) {
  int m = blockIdx.x;
  int tid = threadIdx.x;
  __shared__ float ss[LSER];
  __shared__ int red[LSER];

  ss[tid] = st[m * LSER + tid];
  __syncthreads();

  // bitonic sort ascending, 1024 elements, 1 element per thread
  for (unsigned k = 2; k <= LSER; k <<= 1) {
    for (unsigned j = k >> 1; j > 0; j >>= 1) {
      unsigned ixj = tid ^ j;
      if (ixj > (unsigned)tid) {
        float av = ss[tid], bv = ss[ixj];
        bool up = ((tid & k) == 0);
        bool doswap = up ? (av > bv) : (av < bv);
        if (doswap) { ss[tid] = bv; ss[ixj] = av; }
      }
      __syncthreads();
    }
  }
  float smin = ss[0], smax = ss[LSER - 1];
  float dmax = smax - smin;
  float si = ss[tid];

  auto blockCount = [&](float t) -> int {
    red[tid] = pairCount(ss, si, t);
    __syncthreads();
    for (int off = 512; off > 0; off >>= 1) {
      if (tid < off) red[tid] += red[tid + off];
      __syncthreads();
    }
    int tot = red[0];
    __syncthreads();
    return tot;
  };

  // quantile position = 0.1*(L*L-1) = 104857.5 -> need sorted dists #104857,#104858
  float vres[2];
  for (int pass = 0; pass < 2; ++pass) {
    int target = 104858 + pass;                 // cnt(v) >= target
    int c0 = blockCount(0.0f);
    if (c0 >= target || dmax <= 0.0f) { vres[pass] = 0.0f; continue; }
    unsigned lob = 0u, hib = __float_as_uint(dmax);
    while (hib - lob > 1u) {
      unsigned midb = (lob + hib) >> 1u;
      int cm2 = blockCount(__uint_as_float(midb));
      if (cm2 >= target) hib = midb; else lob = midb;
    }
    vres[pass] = __uint_as_float(hib);
  }
  if (tid == 0) thr[m] = vres[0] + 0.5f * (vres[1] - vres[0]);
}

// ---- fused output: 4-tap bilinear over virtual GAF / RP matrices ----
__global__ void outKernel(const float* __restrict__ c, const float* __restrict__ s,
                          const float* __restrict__ st, const unsigned* __restrict__ gminKey,
                          const unsigned* __restrict__ gmaxKey, const float* __restrict__ thr,
                          float* __restrict__ out) {
  int idx = blockIdx.x * 256 + threadIdx.x;
  if (idx >= OUTN) return;
  int ox = idx % IMG;
  int t1 = idx / IMG;
  int oy = t1 % IMG;
  int t2 = t1 / IMG;
  int ch = t2 % (2 * ND);
  int b  = t2 / (2 * ND);
  int d  = ch & (ND - 1);
  int m  = b * ND + d;

  const float scale = (float)LSER / (float)IMG;
  float sy = ((float)oy + 0.5f) * scale - 0.5f;
  float sx = ((float)ox + 0.5f) * scale - 0.5f;
  float fy = floorf(sy), fx = floorf(sx);
  int y0 = (int)fy, x0 = (int)fx;
  float wy = sy - fy, wx = sx - fx;
  int y0c = min(max(y0, 0), LSER - 1), y1c = min(max(y0 + 1, 0), LSER - 1);
  int x0c = min(max(x0, 0), LSER - 1), x1c = min(max(x0 + 1, 0), LSER - 1);
  int base = m * LSER;

  float v00, v01, v10, v11;
  if (ch < ND) {  // GAF channel
    float gmin = kfloat(gminKey[m]);
    float gmax = kfloat(gmaxKey[m]);
    float rng = gmax - gmin;
    float inv = (rng < EPSN) ? 0.0f : 1.0f / (rng + EPSN);
    float sub = (rng < EPSN) ? 0.0f : gmin;
    float ci0 = c[base + y0c], si0 = s[base + y0c];
    float ci1 = c[base + y1c], si1 = s[base + y1c];
    float cj0 = c[base + x0c], sj0 = s[base + x0c];
    float cj1 = c[base + x1c], sj1 = s[base + x1c];
    v00 = (ci0 * cj0 - si0 * sj0 - sub) * inv;
    v01 = (ci0 * cj1 - si0 * sj1 - sub) * inv;
    v10 = (ci1 * cj0 - si1 * sj0 - sub) * inv;
    v11 = (ci1 * cj1 - si1 * sj1 - sub) * inv;
  } else {        // RP channel
    float th = thr[m];
    float a0 = st[base + y0c], a1 = st[base + y1c];
    float b0 = st[base + x0c], b1 = st[base + x1c];
    v00 = (fabsf(a0 - b0) <= th) ? 1.0f : 0.0f;
    v01 = (fabsf(a0 - b1) <= th) ? 1.0f : 0.0f;
    v10 = (fabsf(a1 - b0) <= th) ? 1.0f : 0.0f;
    v11 = (fabsf(a1 - b1) <= th) ? 1.0f : 0.0f;
  }
  float top = v00 + wx * (v01 - v00);
  float bot = v10 + wx * (v11 - v10);
  out[idx] = top + wy * (bot - top);
}

extern "C" void kernel_launch(void* const* d_in, const int* in_sizes, int n_in,
                              void* d_out, int out_size, void* d_ws, size_t ws_size,
                              hipStream_t stream) {
  const float* x = (const float*)d_in[0];
  float* out = (float*)d_out;

  // workspace layout (floats/uints)
  float* c  = (float*)d_ws;                 // 32768
  float* s  = c + NB * ND * LSER;           // 32768
  float* st = s + NB * ND * LSER;           // 32768
  unsigned* gminKey = (unsigned*)(st + NB * ND * LSER);  // 32
  unsigned* gmaxKey = gminKey + NM;                      // 32
  unsigned* xKeys   = gmaxKey + NM;                      // 2
  float* thr = (float*)(xKeys + 2);                      // 32

  initKernel<<<1, 128, 0, stream>>>(gminKey, gmaxKey, xKeys);
  reduceXKernel<<<32, 1024, 0, stream>>>(x, xKeys);
  prepKernel<<<(NB * ND * LSER) / 256, 256, 0, stream>>>(x, xKeys, c, s, st);
  gafMinMaxKernel<<<dim3(8, NM), 256, 0, stream>>>(c, s, gminKey, gmaxKey);
  rpThrKernel<<<NM, 1024, 0, stream>>>(st, thr);
  outKernel<<<(OUTN + 255) / 256, 256, 0, stream>>>(c, s, st, gminKey, gmaxKey, thr, out);
}